// SoftmaxScatterNormModuleDict_62148176773687
// MI455X (gfx1250) — compile-verified
//
#include <hip/hip_runtime.h>
#include <cmath>

#define BLOCK 256
#define CCH 16
#define QUADS 4
#define TILE_EDGES (BLOCK / QUADS) /* 64 edges per tile (64*64B = 4KB) */
#define EPSV 1e-16f

typedef float v4f __attribute__((ext_vector_type(4)));

// ---------------- CDNA5 async global->LDS path (probe-guarded) -------------
#if defined(__AMDGCN__) && __has_builtin(__builtin_amdgcn_global_load_async_to_lds_b128)
#define HAVE_ASYNC 1
typedef int v4i_vs __attribute__((vector_size(16)));
typedef __attribute__((address_space(1))) v4i_vs g_v4i;
typedef __attribute__((address_space(3))) v4i_vs l_v4i;
__device__ __forceinline__ void async_cp16(const float* g, float* l) {
  __builtin_amdgcn_global_load_async_to_lds_b128((g_v4i*)g, (l_v4i*)l, 0, 0);
}
#if __has_builtin(__builtin_amdgcn_s_wait_asynccnt)
#define WAIT_ASYNC(n) __builtin_amdgcn_s_wait_asynccnt(n)
#else
#define WAIT_ASYNC(n) asm volatile("s_wait_asynccnt %0" ::"i"(n) : "memory")
#endif
#else
#define HAVE_ASYNC 0
#define WAIT_ASYNC(n)
#endif

// ---------------- native L2 float atomics (non-returning, STOREcnt) --------
__device__ __forceinline__ void gmax_f32(float* p, float x) {
  asm volatile("global_atomic_max_num_f32 %0, %1, off scope:SCOPE_DEV"
               :: "v"(p), "v"(x) : "memory");
}
__device__ __forceinline__ void gadd_f32(float* p, float x) {
  asm volatile("global_atomic_add_f32 %0, %1, off scope:SCOPE_DEV"
               :: "v"(p), "v"(x) : "memory");
}

// ---------------- ws layout: [mA | sA | mB | sB], sizes from device scalars -
__global__ __launch_bounds__(BLOCK) void init_ws(float* __restrict__ ws,
                                                 const int* __restrict__ pNA,
                                                 const int* __restrict__ pNB) {
  const long nAC = (long)(*pNA) * CCH;
  const long nBC = (long)(*pNB) * CCH;
  const long total = 2 * nAC + 2 * nBC;
  const long stride = (long)gridDim.x * blockDim.x;
  for (long i = (long)blockIdx.x * blockDim.x + threadIdx.x; i < total; i += stride) {
    float v;
    if (i < nAC)                 v = -__builtin_inff(); // mA
    else if (i < 2 * nAC)        v = 0.0f;              // sA
    else if (i < 2 * nAC + nBC)  v = -__builtin_inff(); // mB
    else                         v = 0.0f;              // sB
    ws[i] = v;
  }
}

// OP: 0 = segment max, 1 = segment sum of exp(x-m), 2 = normalize & write out
template <int OP>
__global__ __launch_bounds__(BLOCK) void edge_pass(const float* __restrict__ fea,
                                                   const int* __restrict__ idx,
                                                   float* __restrict__ ws,
                                                   const int* __restrict__ pNA,
                                                   const int* __restrict__ pNB,
                                                   int which, long E,
                                                   float* __restrict__ out) {
  const long NA = *pNA, NB = *pNB;
  float* base = ws + (which ? 2 * NA * (long)CCH : 0);
  float* __restrict__ m = base;
  float* __restrict__ s = base + (long)(which ? NB : NA) * CCH;

  const int tid  = threadIdx.x;
  const int quad = tid & (QUADS - 1);
  const long numTiles = (E + TILE_EDGES - 1) / TILE_EDGES;
  const long stride = gridDim.x;

#if HAVE_ASYNC
  __shared__ float buf[2][BLOCK * 4]; // 2 x 4KB double buffer, 16B/lane slots
#endif

  long tile = blockIdx.x;
  int cur = 0;
  bool curFull = false;
#if HAVE_ASYNC
  if (tile < numTiles) {
    const long b0 = tile * TILE_EDGES;
    curFull = (b0 + TILE_EDGES) <= E;
    if (curFull) async_cp16(fea + b0 * CCH + tid * 4, &buf[0][tid * 4]);
  }
#endif

  for (; tile < numTiles; tile += stride) {
    const long edge  = tile * TILE_EDGES + (tid >> 2);
    const bool valid = edge < E;
    bool nextFull = false;
#if HAVE_ASYNC
    {
      const long ntile = tile + stride;
      if (ntile < numTiles) {
        const long nb = ntile * TILE_EDGES;
        nextFull = (nb + TILE_EDGES) <= E;
        if (nextFull) async_cp16(fea + nb * CCH + tid * 4, &buf[cur ^ 1][tid * 4]);
      }
    }
#endif
    v4f v;
#if HAVE_ASYNC
    if (curFull) {
      if (nextFull) { WAIT_ASYNC(1); } else { WAIT_ASYNC(0); }
      asm volatile("" ::: "memory");
      v = *(const v4f*)&buf[cur][tid * 4];
    } else
#endif
    if (valid) {
      // streaming read: non-temporal so the edge stream doesn't thrash L2
      v = __builtin_nontemporal_load((const v4f*)(fea + edge * CCH + quad * 4));
    }

    if (valid) {
      const int node = idx[edge];
      float* mp = m + (long)node * CCH + quad * 4;
      if (OP == 0) {
        gmax_f32(mp + 0, v.x);
        gmax_f32(mp + 1, v.y);
        gmax_f32(mp + 2, v.z);
        gmax_f32(mp + 3, v.w);
      } else if (OP == 1) {
        const v4f mv = *(const v4f*)mp; // L2-hot gather
        float* sp = s + (long)node * CCH + quad * 4;
        gadd_f32(sp + 0, expf(v.x - mv.x));
        gadd_f32(sp + 1, expf(v.y - mv.y));
        gadd_f32(sp + 2, expf(v.z - mv.z));
        gadd_f32(sp + 3, expf(v.w - mv.w));
      } else {
        const v4f mv = *(const v4f*)mp;
        const v4f sv = *(const v4f*)(s + (long)node * CCH + quad * 4);
        v4f r;
        r.x = expf(v.x - mv.x) / (sv.x + EPSV);
        r.y = expf(v.y - mv.y) / (sv.y + EPSV);
        r.z = expf(v.z - mv.z) / (sv.z + EPSV);
        r.w = expf(v.w - mv.w) / (sv.w + EPSV);
        __builtin_nontemporal_store(r, (v4f*)(out + edge * CCH + quad * 4));
      }
    }
    cur ^= 1;
    curFull = nextFull;
  }
}

extern "C" void kernel_launch(void* const* d_in, const int* in_sizes, int n_in,
                              void* d_out, int out_size, void* d_ws, size_t ws_size,
                              hipStream_t stream) {
  const float* feaA = (const float*)d_in[0];
  const float* feaB = (const float*)d_in[1];
  const int*   idxA = (const int*)d_in[2];
  const int*   idxB = (const int*)d_in[3];
  const int*   pNA  = (const int*)d_in[4];
  const int*   pNB  = (const int*)d_in[5];

  const long EA = in_sizes[2];
  const long EB = in_sizes[3];

  float* out  = (float*)d_out;
  float* outA = out;
  float* outB = out + EA * (long)CCH;
  float* ws   = (float*)d_ws;

  init_ws<<<2048, BLOCK, 0, stream>>>(ws, pNA, pNB);

  const long tilesA = (EA + TILE_EDGES - 1) / TILE_EDGES;
  const long tilesB = (EB + TILE_EDGES - 1) / TILE_EDGES;
  int gA = (int)(tilesA < 4096 ? (tilesA > 0 ? tilesA : 1) : 4096);
  int gB = (int)(tilesB < 4096 ? (tilesB > 0 ? tilesB : 1) : 4096);

  // pass 1: segment max
  edge_pass<0><<<gA, BLOCK, 0, stream>>>(feaA, idxA, ws, pNA, pNB, 0, EA, nullptr);
  edge_pass<0><<<gB, BLOCK, 0, stream>>>(feaB, idxB, ws, pNA, pNB, 1, EB, nullptr);
  // pass 2: segment sum of exp(x - m)
  edge_pass<1><<<gA, BLOCK, 0, stream>>>(feaA, idxA, ws, pNA, pNB, 0, EA, nullptr);
  edge_pass<1><<<gB, BLOCK, 0, stream>>>(feaB, idxB, ws, pNA, pNB, 1, EB, nullptr);
  // pass 3: normalize, write outputs
  edge_pass<2><<<gA, BLOCK, 0, stream>>>(feaA, idxA, ws, pNA, pNB, 0, EA, outA);
  edge_pass<2><<<gB, BLOCK, 0, stream>>>(feaB, idxB, ws, pNA, pNB, 1, EB, outB);
}